// MyFastPlasticLSTM_14508399526006
// MI455X (gfx1250) — compile-verified
//
#include <hip/hip_runtime.h>
#include <math.h>

#define BB      64
#define IS      512
#define HH      1024
#define FH      4096      // 4*H
#define KTOT    1536      // IS + HH
#define KC      64        // K chunk staged in LDS
#define CLIPV   2.0f

typedef float v2f __attribute__((ext_vector_type(2)));
typedef float v4f __attribute__((ext_vector_type(4)));
typedef float v8f __attribute__((ext_vector_type(8)));

__device__ __forceinline__ float sigmoidf(float x) {
    return 1.0f / (1.0f + __expf(-x));
}

// ---------------------------------------------------------------------------
// K1: allout[b,j] = sum_i X[b,i]*Wx[j,i] + sum_h H0[b,h]*Wh[j,h] + bx[j]+bh[j]
// One block = one 16-wide column tile (j0), 4 waves cover all 64 rows.
// fp32 WMMA 16x16x4, A/B fragments served from LDS, dual accumulators for ILP.
// ---------------------------------------------------------------------------
__global__ __launch_bounds__(128)
void k1_allout_wmma(const float* __restrict__ X,    // B x IS
                    const float* __restrict__ H0,   // B x HH
                    const float* __restrict__ Wx,   // FH x IS
                    const float* __restrict__ Wh,   // FH x HH
                    const float* __restrict__ bx,   // FH
                    const float* __restrict__ bh,   // FH
                    float* __restrict__ allout)     // B x FH
{
    __shared__ float As[64][KC + 4];   // activations: 64 rows x 64 k (pad->bank stride 4)
    __shared__ float Bs[16][KC + 4];   // weights for 16 output cols

    const int tid  = threadIdx.x;
    const int lane = tid & 31;
    const int wave = tid >> 5;
    const int j0   = blockIdx.x * 16;
    const int m0   = wave * 16;
    const int half = lane >> 4;        // 0: lanes 0-15, 1: lanes 16-31
    const int l16  = lane & 15;
    const int klo  = half * 2;         // K sub-offset per ISA A/B fragment layout

    v8f acc0 = {};
    v8f acc1 = {};

    for (int kc = 0; kc < KTOT; kc += KC) {
        const float* asrc; int alda;
        const float* wsrc; int wlda;
        if (kc < IS) { asrc = X  + kc;        alda = IS; wsrc = Wx + kc;        wlda = IS; }
        else         { asrc = H0 + (kc - IS); alda = HH; wsrc = Wh + (kc - IS); wlda = HH; }

        // stage A tile: 64 rows x 64 cols (1024 float4 / 128 threads = 8 each)
        #pragma unroll
        for (int i = 0; i < 8; ++i) {
            int idx = tid + i * 128;
            int r = idx >> 4, c4 = idx & 15;
            v4f v = *reinterpret_cast<const v4f*>(asrc + r * alda + c4 * 4);
            As[r][c4 * 4 + 0] = v.x; As[r][c4 * 4 + 1] = v.y;
            As[r][c4 * 4 + 2] = v.z; As[r][c4 * 4 + 3] = v.w;
        }
        // stage B tile: 16 weight rows x 64 cols
        #pragma unroll
        for (int i = 0; i < 2; ++i) {
            int idx = tid + i * 128;
            int r = idx >> 4, c4 = idx & 15;
            v4f v = *reinterpret_cast<const v4f*>(wsrc + (size_t)(j0 + r) * wlda + c4 * 4);
            Bs[r][c4 * 4 + 0] = v.x; Bs[r][c4 * 4 + 1] = v.y;
            Bs[r][c4 * 4 + 2] = v.z; Bs[r][c4 * 4 + 3] = v.w;
        }
        __syncthreads();

        #pragma unroll
        for (int ks = 0; ks < KC; ks += 8) {
            v2f a0, b0, a1, b1;
            // A 16x4 fragment: lane half selects K pair {0,1} vs {2,3}
            a0.x = As[m0 + l16][ks + klo];
            a0.y = As[m0 + l16][ks + klo + 1];
            b0.x = Bs[l16][ks + klo];
            b0.y = Bs[l16][ks + klo + 1];
            a1.x = As[m0 + l16][ks + 4 + klo];
            a1.y = As[m0 + l16][ks + 4 + klo + 1];
            b1.x = Bs[l16][ks + 4 + klo];
            b1.y = Bs[l16][ks + 4 + klo + 1];
            acc0 = __builtin_amdgcn_wmma_f32_16x16x4_f32(
                false, a0, false, b0, (short)0, acc0, false, false);
            acc1 = __builtin_amdgcn_wmma_f32_16x16x4_f32(
                false, a1, false, b1, (short)0, acc1, false, false);
        }
        __syncthreads();
    }

    // C/D layout: VGPR r -> row m0 + r + 8*half, col j0 + l16
    const int col  = j0 + l16;
    const float bias = bx[col] + bh[col];
    #pragma unroll
    for (int r = 0; r < 8; ++r) {
        int row = m0 + half * 8 + r;
        allout[(size_t)row * FH + col] = acc0[r] + acc1[r] + bias;
    }
}

// ---------------------------------------------------------------------------
// K2: hebbterm + gates + cell + hactiv (first 256MB hebb read, coalesced in k).
// Default (RT) temporal hint on hebb so the tail batches stay in the 192MB L2
// for K4's reverse-order re-read. grid: (HH/256, BB) -> b on the slow axis, so
// high batches are read last (hot in L2 when K2 finishes).
// ---------------------------------------------------------------------------
__global__ __launch_bounds__(256)
void k2_hebbterm_gates(const float* __restrict__ H0,
                       const float* __restrict__ C0,
                       const float* __restrict__ alpha,   // HH x HH
                       const float* __restrict__ hebb,    // BB x HH x HH
                       const float* __restrict__ allout,  // BB x FH
                       float* __restrict__ inputtoc,      // ws: BB x HH
                       float* __restrict__ hact,          // out[0]
                       float* __restrict__ cellout)       // out[1]
{
    __shared__ float h0s[HH];
    const int b = blockIdx.y;
    const int k = blockIdx.x * 256 + threadIdx.x;

    #pragma unroll
    for (int i = 0; i < 4; ++i)
        h0s[threadIdx.x + 256 * i] = H0[b * HH + threadIdx.x + 256 * i];
    __syncthreads();

    const float* hb = hebb + (size_t)b * HH * HH + k;
    const float* al = alpha + k;
    float tacc = 0.0f;
    #pragma unroll 4
    for (int h = 0; h < HH; ++h) {
        tacc = fmaf(h0s[h] * al[(size_t)h * HH], hb[(size_t)h * HH], tacc);
    }

    const float* ao = allout + (size_t)b * FH;
    float fgt  = sigmoidf(ao[k]);
    float ipt  = sigmoidf(ao[HH + k]);
    float opt  = sigmoidf(ao[2 * HH + k]);
    float itc  = tanhf(ao[3 * HH + k] + tacc);
    float cell = fgt * C0[b * HH + k] + ipt * itc;
    float ha   = opt * tanhf(cell);

    inputtoc[b * HH + k] = itc;
    hact[b * HH + k]     = ha;
    cellout[b * HH + k]  = cell;
}

// ---------------------------------------------------------------------------
// K3: m[b] = tanh(hactiv[b,:]·Wm + bm); g[b,k] = (m*Wf[k]+bf[k]) * inputtoc[b,k]
// grid: BB blocks of 256.
// ---------------------------------------------------------------------------
__global__ __launch_bounds__(256)
void k3_modulator(const float* __restrict__ hact,
                  const float* __restrict__ Wm, const float* __restrict__ bm,
                  const float* __restrict__ Wf, const float* __restrict__ bf,
                  const float* __restrict__ inputtoc,
                  float* __restrict__ g)              // ws: BB x HH
{
    __shared__ float red[256];
    __shared__ float mval;
    const int b = blockIdx.x;

    float p = 0.0f;
    #pragma unroll
    for (int i = 0; i < 4; ++i) {
        int k = threadIdx.x + 256 * i;
        p = fmaf(hact[b * HH + k], Wm[k], p);
    }
    red[threadIdx.x] = p;
    __syncthreads();
    for (int s = 128; s > 0; s >>= 1) {
        if (threadIdx.x < s) red[threadIdx.x] += red[threadIdx.x + s];
        __syncthreads();
    }
    if (threadIdx.x == 0) mval = tanhf(red[0] + bm[0]);
    __syncthreads();
    const float m = mval;

    #pragma unroll
    for (int i = 0; i < 4; ++i) {
        int k = threadIdx.x + 256 * i;
        g[b * HH + k] = fmaf(m, Wf[k], bf[k]) * inputtoc[b * HH + k];
    }
}

// ---------------------------------------------------------------------------
// K4: hebb_new = clip(hebb + h0[b,h]*g[b,k], +-2).
// REVERSE block order: start with the high-b batches that K2 touched last and
// that are still resident in the 192MB L2 -> a large share of this second
// 256MB read hits L2 instead of HBM. NT (last-use) load on hebb since the
// data is dead afterwards; NT store on hebb_new (never re-read) so the write
// stream does not evict the cached hebb lines.
// grid: BB*HH blocks (one (b,h) row each), block 256 lanes x float4.
// ---------------------------------------------------------------------------
__global__ __launch_bounds__(256)
void k4_hebb_update(const float* __restrict__ hebb,
                    const float* __restrict__ H0,
                    const float* __restrict__ g,     // BB x HH (L2 resident)
                    float* __restrict__ hebbn)       // out[2]
{
    const int row = (BB * HH - 1) - blockIdx.x;   // walk batches high -> low
    const int b   = row >> 10;
    const float hh = H0[row];

    const size_t base = (size_t)row * HH + (size_t)threadIdx.x * 4;
    const v4f* src = reinterpret_cast<const v4f*>(hebb + base);
    v4f*       dst = reinterpret_cast<v4f*>(hebbn + base);
    v4f gv  = *reinterpret_cast<const v4f*>(g + b * HH + threadIdx.x * 4);
    v4f old = __builtin_nontemporal_load(src);

    v4f nv;
    nv.x = fminf(fmaxf(fmaf(hh, gv.x, old.x), -CLIPV), CLIPV);
    nv.y = fminf(fmaxf(fmaf(hh, gv.y, old.y), -CLIPV), CLIPV);
    nv.z = fminf(fmaxf(fmaf(hh, gv.z, old.z), -CLIPV), CLIPV);
    nv.w = fminf(fmaxf(fmaf(hh, gv.w, old.w), -CLIPV), CLIPV);
    __builtin_nontemporal_store(nv, dst);
}

// ---------------------------------------------------------------------------
extern "C" void kernel_launch(void* const* d_in, const int* in_sizes, int n_in,
                              void* d_out, int out_size, void* d_ws, size_t ws_size,
                              hipStream_t stream) {
    const float* X     = (const float*)d_in[0];   // inputs  B x IS
    const float* H0    = (const float*)d_in[1];   // h0      B x HH
    const float* C0    = (const float*)d_in[2];   // c0      B x HH
    const float* hebb  = (const float*)d_in[3];   // B x HH x HH
    const float* Wx    = (const float*)d_in[4];   // FH x IS
    const float* bx    = (const float*)d_in[5];   // FH
    const float* Wh    = (const float*)d_in[6];   // FH x HH
    const float* bh    = (const float*)d_in[7];   // FH
    const float* alpha = (const float*)d_in[8];   // HH x HH
    const float* Wm    = (const float*)d_in[9];   // 1 x HH
    const float* bm    = (const float*)d_in[10];  // 1
    const float* Wf    = (const float*)d_in[11];  // HH x 1
    const float* bf    = (const float*)d_in[12];  // HH

    float* out   = (float*)d_out;
    float* hact  = out;                     // B x HH
    float* cell  = out + BB * HH;           // B x HH
    float* hebbn = out + 2 * BB * HH;       // B x HH x HH

    float* ws       = (float*)d_ws;
    float* allout   = ws;                   // B*FH    (1 MB)
    float* inputtoc = ws + BB * FH;         // B*HH    (256 KB)
    float* g        = inputtoc + BB * HH;   // B*HH    (256 KB)
    (void)ws_size; (void)in_sizes; (void)n_in; (void)out_size;

    k1_allout_wmma<<<dim3(FH / 16), 128, 0, stream>>>(X, H0, Wx, Wh, bx, bh, allout);
    k2_hebbterm_gates<<<dim3(HH / 256, BB), 256, 0, stream>>>(H0, C0, alpha, hebb,
                                                              allout, inputtoc, hact, cell);
    k3_modulator<<<dim3(BB), 256, 0, stream>>>(hact, Wm, bm, Wf, bf, inputtoc, g);
    k4_hebb_update<<<dim3(BB * HH), 256, 0, stream>>>(hebb, H0, g, hebbn);
}